// EpsilonNetGM_16269336118066
// MI455X (gfx1250) — compile-verified
//
#include <hip/hip_runtime.h>
#include <math.h>

typedef __attribute__((ext_vector_type(2))) float v2f;
typedef __attribute__((ext_vector_type(8))) float v8f;

#define KC 16
#define DD 64
#define WAVES 4

// workspace layout (floats)
#define WS_LROW 0                      // Linv row-major:  [k][i][j] = Linv[i][j]   (B of GEMM1)
#define WS_LCOL (KC*DD*DD)             // [k][n][j] = Linv[j][n]                    (B of GEMM2)
#define WS_MU   (2*KC*DD*DD)           // [k][d] = sqrt(acp)*means
#define WS_CK   (WS_MU + KC*DD)        // [k] log-weight + const
#define WS_SCALE (WS_CK + KC)          // sqrt(1-acp)

// ---------------------------------------------------------------------------
// Setup: per-k Cholesky of (1-acp)I + acp*C, Linv via forward substitution.
// ---------------------------------------------------------------------------
__global__ __launch_bounds__(64) void gm_setup_kernel(
    const float* __restrict__ means, const float* __restrict__ weights,
    const float* __restrict__ covs,  const float* __restrict__ acp,
    const int* __restrict__ tptr, float* __restrict__ ws) {
  __shared__ float A[DD][DD];   // sigma -> cholesky L (lower)
  __shared__ float X[DD][DD];   // Linv (lower)
  const int k = blockIdx.x;
  const int tid = threadIdx.x;          // 64 threads
  const float a = acp[tptr[0]];

  for (int i = 0; i < DD; ++i) {
    float s = a * covs[(k * DD + i) * DD + tid];
    if (i == tid) s += (1.0f - a);
    A[i][tid] = s;
    X[i][tid] = 0.0f;
  }
  __syncthreads();

  // left-looking Cholesky, one column per outer step
  for (int j = 0; j < DD; ++j) {
    if (tid == j) {
      float s = A[j][j];
      for (int p = 0; p < j; ++p) s -= A[j][p] * A[j][p];
      A[j][j] = sqrtf(s);
    }
    __syncthreads();
    if (tid > j) {
      float s = A[tid][j];
      for (int p = 0; p < j; ++p) s -= A[tid][p] * A[j][p];
      A[tid][j] = s / A[j][j];
    }
    __syncthreads();
  }

  // Linv: solve L * X = I, one column per thread
  {
    const int c = tid;
    X[c][c] = 1.0f / A[c][c];
    for (int i = c + 1; i < DD; ++i) {
      float s = 0.0f;
      for (int p = c; p < i; ++p) s += A[i][p] * X[p][c];
      X[i][c] = -s / A[i][i];
    }
  }
  __syncthreads();

  for (int j = 0; j < DD; ++j) {
    ws[WS_LROW + (k * DD + tid) * DD + j] = X[tid][j];   // Linv[i=tid][j]
    ws[WS_LCOL + (k * DD + tid) * DD + j] = X[j][tid];   // Linv[j][n=tid]
  }
  ws[WS_MU + k * DD + tid] = sqrtf(a) * means[k * DD + tid];

  if (tid == 0) {
    float logdet = 0.0f;
    for (int i = 0; i < DD; ++i) logdet += logf(A[i][i]);
    logdet *= 2.0f;
    float wsum = 0.0f;
    for (int kk = 0; kk < KC; ++kk) wsum += weights[kk];
    const float LOG2PI = 1.8378770664093453f;
    ws[WS_CK + k] = logf(weights[k]) - logf(wsum)
                  - 0.5f * ((float)DD * LOG2PI + logdet);
    if (k == 0) ws[WS_SCALE] = sqrtf(1.0f - a);
  }
}

// ---------------------------------------------------------------------------
// Main: per wave32, 16 samples. Two fp32 WMMA GEMMs per component with an
// online softmax across the K=16 components.
// ---------------------------------------------------------------------------
__global__ __launch_bounds__(WAVES * 32) void gm_score_kernel(
    const float* __restrict__ x, const float* __restrict__ ws,
    float* __restrict__ out) {
  __shared__ float sMu[KC * DD];
  __shared__ float sCk[KC];
  __shared__ float sY[WAVES][16 * DD];  // per-wave Y tile (D-layout -> A-layout bounce)
  __shared__ float sW[WAVES][16];       // per-wave per-row softmax numerator weight

  const int tid = threadIdx.x;
  for (int i = tid; i < KC * DD; i += WAVES * 32) sMu[i] = ws[WS_MU + i];
  if (tid < KC) sCk[tid] = ws[WS_CK + tid];
  __syncthreads();

  const int wave = tid >> 5;
  const int lane = tid & 31;
  const int h    = lane >> 4;   // lane half: selects K-subpair in A/B frags, row-half in C/D
  const int lr   = lane & 15;
  const int m0   = (blockIdx.x * WAVES + wave) * 16;

  const float* __restrict__ Lrow = ws + WS_LROW;
  const float* __restrict__ Lcol = ws + WS_LCOL;
  const float scaleFac = ws[WS_SCALE];

  // x tile in A-fragment layout: lane holds x[m0+lr][4*kb + 2h + {0,1}]
  v2f xf[16];
#pragma unroll
  for (int kb = 0; kb < 16; ++kb)
    xf[kb] = *(const v2f*)(x + (m0 + lr) * DD + 4 * kb + 2 * h);

  v8f G[4] = {};
  float M[8], Z[8];
#pragma unroll
  for (int r = 0; r < 8; ++r) { M[r] = -INFINITY; Z[r] = 0.0f; }

  for (int k = 0; k < KC; ++k) {
    const float ck = sCk[k];
    // ---- GEMM1: Y(16x64) = (x - mu) @ Linv^T ----
    v8f acc[4] = {};
#pragma unroll
    for (int kb = 0; kb < 16; ++kb) {
      v2f mu = *(const v2f*)&sMu[k * DD + 4 * kb + 2 * h];
      v2f af; af.x = xf[kb].x - mu.x; af.y = xf[kb].y - mu.y;
#pragma unroll
      for (int nb = 0; nb < 4; ++nb) {
        v2f bf = *(const v2f*)(Lrow + ((k * DD + nb * 16 + lr) * DD + 4 * kb + 2 * h));
        acc[nb] = __builtin_amdgcn_wmma_f32_16x16x4_f32(
            false, af, false, bf, (short)0, acc[nb], false, false);
      }
    }

    // ---- mahalanobis per row: sum over n of Y^2, butterfly within lane half ----
    float wrow[8];
#pragma unroll
    for (int r = 0; r < 8; ++r) {
      float s = acc[0][r] * acc[0][r] + acc[1][r] * acc[1][r]
              + acc[2][r] * acc[2][r] + acc[3][r] * acc[3][r];
      s += __shfl_xor(s, 1, 32);
      s += __shfl_xor(s, 2, 32);
      s += __shfl_xor(s, 4, 32);
      s += __shfl_xor(s, 8, 32);
      // online softmax update for row m = 8h + r
      float logit = ck - 0.5f * s;
      float nm    = fmaxf(M[r], logit);
      float alpha = __expf(M[r] - nm);
      float wk    = __expf(logit - nm);
      Z[r] = Z[r] * alpha + wk;
      M[r] = nm;
      wrow[r] = wk;
#pragma unroll
      for (int nb = 0; nb < 4; ++nb) G[nb][r] *= alpha;  // rescale accumulator
    }

    // ---- bounce Y through LDS: C/D layout -> row-major -> A fragments ----
#pragma unroll
    for (int nb = 0; nb < 4; ++nb)
#pragma unroll
      for (int r = 0; r < 8; ++r)
        sY[wave][(8 * h + r) * DD + nb * 16 + lr] = acc[nb][r];
    if (lr == 0) {
#pragma unroll
      for (int r = 0; r < 8; ++r) sW[wave][8 * h + r] = wrow[r];
    }
    asm volatile("s_wait_dscnt 0" ::: "memory");   // wave-private region, cross-lane read next
    __builtin_amdgcn_wave_barrier();

    const float wA = sW[wave][lr];                 // weight for A-frag row m = lr
    // ---- GEMM2: G += (w .* Y) @ Linv ----
#pragma unroll
    for (int kb = 0; kb < 16; ++kb) {
      v2f y2 = *(const v2f*)&sY[wave][lr * DD + 4 * kb + 2 * h];
      v2f af; af.x = y2.x * wA; af.y = y2.y * wA;
#pragma unroll
      for (int nb = 0; nb < 4; ++nb) {
        v2f bf = *(const v2f*)(Lcol + ((k * DD + nb * 16 + lr) * DD + 4 * kb + 2 * h));
        G[nb] = __builtin_amdgcn_wmma_f32_16x16x4_f32(
            false, af, false, bf, (short)0, G[nb], false, false);
      }
    }
    __builtin_amdgcn_wave_barrier();               // keep next iter's sY stores behind these loads
  }

  // out = sqrt(1-acp) * G / Z
#pragma unroll
  for (int nb = 0; nb < 4; ++nb)
#pragma unroll
    for (int r = 0; r < 8; ++r)
      out[(m0 + 8 * h + r) * DD + nb * 16 + lr] = scaleFac * G[nb][r] / Z[r];
}

// ---------------------------------------------------------------------------
extern "C" void kernel_launch(void* const* d_in, const int* in_sizes, int n_in,
                              void* d_out, int out_size, void* d_ws, size_t ws_size,
                              hipStream_t stream) {
  const float* x       = (const float*)d_in[0];
  const float* means   = (const float*)d_in[1];
  const float* weights = (const float*)d_in[2];
  const float* covs    = (const float*)d_in[3];
  const float* acp     = (const float*)d_in[4];
  const int*   t       = (const int*)d_in[5];
  float* out = (float*)d_out;
  float* ws  = (float*)d_ws;

  const int B = in_sizes[0] / DD;          // 65536

  gm_setup_kernel<<<KC, 64, 0, stream>>>(means, weights, covs, acp, t, ws);
  gm_score_kernel<<<B / (WAVES * 16), WAVES * 32, 0, stream>>>(x, ws, out);
}